// GCNExtractor_5669356835492
// MI455X (gfx1250) — compile-verified
//
#include <hip/hip_runtime.h>
#include <hip/hip_bf16.h>

#define NN 2048
#define FF 128
#define KEEP 1258291u

typedef float v2f __attribute__((ext_vector_type(2)));
typedef float v8f __attribute__((ext_vector_type(8)));

struct RadixState {
    unsigned prefix;
    unsigned remaining;
    unsigned pad[2];
    unsigned hist[256];
};

// monotone float -> uint key (larger float => larger key)
__device__ __forceinline__ unsigned fkey(float f) {
    unsigned u = __float_as_uint(f);
    return (u & 0x80000000u) ? ~u : (u | 0x80000000u);
}

// ---- WMMA helpers: V_WMMA_F32_16X16X4_F32, ISA 7.12.2 layouts ----
// A 16x4: lane L holds row m=L&15, VGPR0=K(2*(L>>4)), VGPR1=K(2*(L>>4)+1)
// B 4x16: lane L holds col n=L&15, VGPR0=K(2*(L>>4)), VGPR1=K(2*(L>>4)+1)
// C/D 16x16: VGPR v, lane L -> M = v + 8*(L>>4), N = L&15

// B tile stored N-major: Bt[n][k]  (used when B source is row-per-n, e.g. X X^T)
__device__ __forceinline__ void mma_chunk_bt(const float (*At)[36],
                                             const float (*Bt)[36],
                                             int wm, int wn, int lane, v8f& acc) {
    const int mn = lane & 15;
    const int kh = (lane >> 4) << 1;  // 0 or 2
#pragma unroll
    for (int kk = 0; kk < 32; kk += 4) {
        v2f a, b;
        a.x = At[wm + mn][kk + kh];
        a.y = At[wm + mn][kk + kh + 1];
        b.x = Bt[wn + mn][kk + kh];
        b.y = Bt[wn + mn][kk + kh + 1];
        acc = __builtin_amdgcn_wmma_f32_16x16x4_f32(
            false, a, false, b, (short)0, acc, false, false);
    }
}

// B tile stored K-major: Bk[k][n]  (used when B source is row-major [K][N])
__device__ __forceinline__ void mma_chunk_bk(const float (*At)[36],
                                             const float (*Bk)[40],
                                             int wm, int wn, int lane, v8f& acc) {
    const int mn = lane & 15;
    const int kh = (lane >> 4) << 1;  // 0 or 2
#pragma unroll
    for (int kk = 0; kk < 32; kk += 4) {
        v2f a, b;
        a.x = At[wm + mn][kk + kh];
        a.y = At[wm + mn][kk + kh + 1];
        b.x = Bk[kk + kh][wn + mn];
        b.y = Bk[kk + kh + 1][wn + mn];
        acc = __builtin_amdgcn_wmma_f32_16x16x4_f32(
            false, a, false, b, (short)0, acc, false, false);
    }
}

// ---------------- S = X * X^T - I  (2048x2048, K=128) ----------------
__global__ __launch_bounds__(128) void sim_gemm(const float* __restrict__ X,
                                                float* __restrict__ S) {
    __shared__ float At[32][36];
    __shared__ float Bt[32][36];
    const int tid = threadIdx.x;
    const int wave = tid >> 5, lane = tid & 31;
    const int bm = blockIdx.x * 32, bn = blockIdx.y * 32;
    const int wm = (wave >> 1) * 16, wn = (wave & 1) * 16;
    v8f acc = {0.f, 0.f, 0.f, 0.f, 0.f, 0.f, 0.f, 0.f};

    for (int k0 = 0; k0 < FF; k0 += 32) {
#pragma unroll
        for (int l = 0; l < 2; ++l) {            // 256 float4 slots / 128 threads
            int t = tid + l * 128;
            int i = t >> 3, k = (t & 7) * 4;
            *(float4*)&At[i][k] = *(const float4*)&X[(bm + i) * FF + k0 + k];
            *(float4*)&Bt[i][k] = *(const float4*)&X[(bn + i) * FF + k0 + k];
        }
        __syncthreads();
        mma_chunk_bt(At, Bt, wm, wn, lane, acc);
        __syncthreads();
    }
    const int n = bn + wn + (lane & 15);
    const int mb = bm + wm + ((lane >> 4) << 3);
#pragma unroll
    for (int v = 0; v < 8; ++v) {
        int m = mb + v;
        float val = acc[v];
        if (m == n) val -= 1.0f;  // sim = X X^T - I
        S[m * NN + n] = val;
    }
}

// ---------------- radix-select threshold over 4.2M floats ----------------
__global__ void radix_init(RadixState* st) {
    if (threadIdx.x == 0) {
        st->prefix = 0u;
        st->remaining = KEEP;
    }
    st->hist[threadIdx.x] = 0u;
}

__global__ __launch_bounds__(256) void radix_hist(const float* __restrict__ S,
                                                  RadixState* st, int pass) {
    __shared__ unsigned h[256];
    h[threadIdx.x] = 0u;
    __syncthreads();
    const unsigned prefix = st->prefix;
    const int shift = 24 - 8 * pass;
    const float4* S4 = (const float4*)S;
#pragma unroll
    for (int i = 0; i < 4; ++i) {
        int e4 = blockIdx.x * 1024 + i * 256 + threadIdx.x;
        float4 f = S4[e4];
        unsigned k0 = fkey(f.x), k1 = fkey(f.y), k2 = fkey(f.z), k3 = fkey(f.w);
        if (pass == 0) {
            atomicAdd(&h[(k0 >> shift) & 255u], 1u);
            atomicAdd(&h[(k1 >> shift) & 255u], 1u);
            atomicAdd(&h[(k2 >> shift) & 255u], 1u);
            atomicAdd(&h[(k3 >> shift) & 255u], 1u);
        } else {
            if ((k0 >> (shift + 8)) == prefix) atomicAdd(&h[(k0 >> shift) & 255u], 1u);
            if ((k1 >> (shift + 8)) == prefix) atomicAdd(&h[(k1 >> shift) & 255u], 1u);
            if ((k2 >> (shift + 8)) == prefix) atomicAdd(&h[(k2 >> shift) & 255u], 1u);
            if ((k3 >> (shift + 8)) == prefix) atomicAdd(&h[(k3 >> shift) & 255u], 1u);
        }
    }
    __syncthreads();
    unsigned c = h[threadIdx.x];
    if (c) atomicAdd(&st->hist[threadIdx.x], c);
}

__global__ void radix_scan(RadixState* st) {
    unsigned rem = st->remaining;
    unsigned pre = st->prefix;
    unsigned newpre = pre << 8;
    for (int b = 255; b >= 0; --b) {
        unsigned c = st->hist[b];
        if (rem > c) {
            rem -= c;
        } else {
            newpre = (pre << 8) | (unsigned)b;
            break;
        }
    }
    st->prefix = newpre;
    st->remaining = rem;
    for (int i = 0; i < 256; ++i) st->hist[i] = 0u;
}

// ------- A = select(S, key >= t) + I  (in place; symmetric) -------
__global__ __launch_bounds__(256) void select_kernel(float* __restrict__ S,
                                                     const RadixState* st) {
    const unsigned tkey = st->prefix;
    float4* S4 = (float4*)S;
#pragma unroll
    for (int i = 0; i < 4; ++i) {
        int e4 = blockIdx.x * 1024 + i * 256 + threadIdx.x;
        float4 f = S4[e4];
        int e = e4 * 4;
        int r = e >> 11;
        int c = e & (NN - 1);  // c..c+3 within one row (NN multiple of 4)
        float4 v;
        v.x = (fkey(f.x) >= tkey) ? f.x : 0.0f;
        v.y = (fkey(f.y) >= tkey) ? f.y : 0.0f;
        v.z = (fkey(f.z) >= tkey) ? f.z : 0.0f;
        v.w = (fkey(f.w) >= tkey) ? f.w : 0.0f;
        if (r == c + 0) v.x += 1.0f;  // self-loop weight 1.0 on diagonal
        if (r == c + 1) v.y += 1.0f;
        if (r == c + 2) v.z += 1.0f;
        if (r == c + 3) v.w += 1.0f;
        S4[e4] = v;
    }
}

// ------- deg[r] = row sum of A (== col sum, A symmetric); dinv = rsqrt -------
__global__ __launch_bounds__(256) void deg_kernel(const float* __restrict__ A,
                                                  float* __restrict__ dinv) {
    __shared__ float red[256];
    const int r = blockIdx.x;
    const float4* A4 = (const float4*)(A + (size_t)r * NN);
    float s = 0.f;
    for (int c4 = threadIdx.x; c4 < NN / 4; c4 += 256) {
        float4 f = A4[c4];
        s += (f.x + f.y) + (f.z + f.w);
    }
    red[threadIdx.x] = s;
    __syncthreads();
    for (int off = 128; off > 0; off >>= 1) {
        if (threadIdx.x < off) red[threadIdx.x] += red[threadIdx.x + off];
        __syncthreads();
    }
    if (threadIdx.x == 0) {
        float d = red[0];
        dinv[r] = (d > 0.f) ? rsqrtf(d) : 0.f;
    }
}

// ---------------- h2 = dinv[r] * (X @ W)   (2048x128, K=128) ----------------
__global__ __launch_bounds__(128) void h_gemm(const float* __restrict__ X,
                                              const float* __restrict__ W,
                                              const float* __restrict__ dinv,
                                              float* __restrict__ h2) {
    __shared__ float At[32][36];
    __shared__ float Bk[32][40];
    const int tid = threadIdx.x;
    const int wave = tid >> 5, lane = tid & 31;
    const int bm = blockIdx.x * 32, bn = blockIdx.y * 32;
    const int wm = (wave >> 1) * 16, wn = (wave & 1) * 16;
    v8f acc = {0.f, 0.f, 0.f, 0.f, 0.f, 0.f, 0.f, 0.f};

    for (int k0 = 0; k0 < FF; k0 += 32) {
#pragma unroll
        for (int l = 0; l < 2; ++l) {
            int t = tid + l * 128;
            int i = t >> 3, k = (t & 7) * 4;
            *(float4*)&At[i][k] = *(const float4*)&X[(bm + i) * FF + k0 + k];
            // Bk[k][n] = W[k0+k][bn+n]  (row-major source -> straight vector copy)
            *(float4*)&Bk[i][k] = *(const float4*)&W[(k0 + i) * FF + bn + k];
        }
        __syncthreads();
        mma_chunk_bk(At, Bk, wm, wn, lane, acc);
        __syncthreads();
    }
    const int n = bn + wn + (lane & 15);
    const int mb = bm + wm + ((lane >> 4) << 3);
#pragma unroll
    for (int v = 0; v < 8; ++v) {
        int m = mb + v;
        h2[m * FF + n] = dinv[m] * acc[v];
    }
}

// -------- out = dinv[r] * (A @ h2) + b   (2048x128, K=2048) --------
__global__ __launch_bounds__(128) void out_gemm(const float* __restrict__ A,
                                                const float* __restrict__ h2,
                                                const float* __restrict__ dinv,
                                                const float* __restrict__ bias,
                                                float* __restrict__ out) {
    __shared__ float At[32][36];
    __shared__ float Bk[32][40];
    const int tid = threadIdx.x;
    const int wave = tid >> 5, lane = tid & 31;
    const int bm = blockIdx.x * 32, bn = blockIdx.y * 32;
    const int wm = (wave >> 1) * 16, wn = (wave & 1) * 16;
    v8f acc = {0.f, 0.f, 0.f, 0.f, 0.f, 0.f, 0.f, 0.f};

    for (int k0 = 0; k0 < NN; k0 += 32) {
#pragma unroll
        for (int l = 0; l < 2; ++l) {
            int t = tid + l * 128;
            int i = t >> 3, k = (t & 7) * 4;
            *(float4*)&At[i][k] = *(const float4*)&A[(size_t)(bm + i) * NN + k0 + k];
            // Bk[k][n] = h2[k0+k][bn+n]
            *(float4*)&Bk[i][k] = *(const float4*)&h2[(k0 + i) * FF + bn + k];
        }
        __syncthreads();
        mma_chunk_bk(At, Bk, wm, wn, lane, acc);
        __syncthreads();
    }
    const int n = bn + wn + (lane & 15);
    const int mb = bm + wm + ((lane >> 4) << 3);
    const float bn_val = bias[n];
#pragma unroll
    for (int v = 0; v < 8; ++v) {
        int m = mb + v;
        out[m * FF + n] = dinv[m] * acc[v] + bn_val;
    }
}

extern "C" void kernel_launch(void* const* d_in, const int* in_sizes, int n_in,
                              void* d_out, int out_size, void* d_ws, size_t ws_size,
                              hipStream_t stream) {
    const float* X = (const float*)d_in[0];  // [1,2048,128]
    const float* W = (const float*)d_in[1];  // [128,128]
    const float* b = (const float*)d_in[2];  // [128]
    float* out = (float*)d_out;              // [1,2048,128]

    char* ws = (char*)d_ws;
    float* S    = (float*)(ws);                                    // 16 MB: sim / A
    float* h2   = (float*)(ws + (size_t)NN * NN * 4);              // 1 MB
    float* dinv = (float*)(ws + (size_t)NN * NN * 4 + (size_t)NN * FF * 4);  // 8 KB
    RadixState* st = (RadixState*)(ws + (size_t)NN * NN * 4 + (size_t)NN * FF * 4 + (size_t)NN * 4);

    // 1) S = X X^T - I
    sim_gemm<<<dim3(NN / 32, NN / 32), 128, 0, stream>>>(X, S);

    // 2) radix-select the KEEP-th largest value (exact key) over N^2 entries
    radix_init<<<1, 256, 0, stream>>>(st);
    for (int pass = 0; pass < 4; ++pass) {
        radix_hist<<<(NN * NN) / 4096, 256, 0, stream>>>(S, st, pass);
        radix_scan<<<1, 1, 0, stream>>>(st);
    }

    // 3) A = select(S) + I (in place, stays symmetric)
    select_kernel<<<(NN * NN) / 4096, 256, 0, stream>>>(S, st);

    // 4) deg / dinv via coalesced row sums (A symmetric => row sum == col sum)
    deg_kernel<<<NN, 256, 0, stream>>>(S, dinv);

    // 5) h2 = dinv ⊙ (X @ W)
    h_gemm<<<dim3(NN / 32, FF / 32), 128, 0, stream>>>(X, W, dinv, h2);

    // 6) out = dinv ⊙ (A @ h2) + b
    out_gemm<<<dim3(NN / 32, FF / 32), 128, 0, stream>>>(S, h2, dinv, b, out);
}